// GridPooling_11244224381389
// MI455X (gfx1250) — compile-verified
//
#include <hip/hip_runtime.h>

typedef __attribute__((ext_vector_type(2))) float v2f;
typedef __attribute__((ext_vector_type(8))) float v8f;

#define CIN    64
#define COUT   128
#define KPOOL  128                                  // GRID/STRIDE
#define KEYBITS 21                                  // log2(KPOOL^3)
#define KEYSPACE (4 * KPOOL * KPOOL * KPOOL)        // NB * K^3 = 1<<23
#define SCAN_BLOCK 256
#define SCAN_ITEMS 8
#define SCAN_TILE (SCAN_BLOCK * SCAN_ITEMS)         // 2048
#define NUM_SCAN_BLOCKS (KEYSPACE / SCAN_TILE)      // 4096

// ---------------------------------------------------------------- utilities
__global__ void gp_zero(unsigned* __restrict__ p, size_t n) {
  size_t i = (size_t)blockIdx.x * blockDim.x + threadIdx.x;
  size_t stride = (size_t)gridDim.x * blockDim.x;
  for (; i < n; i += stride) p[i] = 0u;
}

// ------------------------------------------------- key computation + histogram
__global__ void gp_keys_count(const int* __restrict__ gc, const int* __restrict__ batch,
                              int* __restrict__ keys, unsigned* __restrict__ cnt, int n) {
  int i = blockIdx.x * blockDim.x + threadIdx.x;
  if (i >= n) return;
  int gx = gc[3 * i] >> 1, gy = gc[3 * i + 1] >> 1, gz = gc[3 * i + 2] >> 1;
  int key = ((batch[i] * KPOOL + gx) * KPOOL + gy) * KPOOL + gz;
  keys[i] = key;
  atomicAdd(&cnt[key], 1u);
}

// -------------------------------- hierarchical exclusive scan of occupancy bits
__global__ __launch_bounds__(SCAN_BLOCK)
void gp_scan_pass1(const unsigned* __restrict__ cnt, unsigned* __restrict__ blockSums) {
  __shared__ unsigned tmp[SCAN_BLOCK];
  const int t = threadIdx.x;
  size_t base = (size_t)blockIdx.x * SCAN_TILE + (size_t)t * SCAN_ITEMS;
  unsigned s = 0;
#pragma unroll
  for (int i = 0; i < SCAN_ITEMS; ++i) s += (cnt[base + i] != 0u) ? 1u : 0u;
  tmp[t] = s;
  __syncthreads();
  for (int off = SCAN_BLOCK / 2; off > 0; off >>= 1) {
    if (t < off) tmp[t] += tmp[t + off];
    __syncthreads();
  }
  if (t == 0) blockSums[blockIdx.x] = tmp[0];
}

__global__ __launch_bounds__(256)
void gp_scan_pass2(unsigned* __restrict__ blockSums, unsigned* __restrict__ uniqCount) {
  __shared__ unsigned tmp[256];
  const int t = threadIdx.x;
  unsigned loc[16];
  unsigned s = 0;
#pragma unroll
  for (int i = 0; i < 16; ++i) { loc[i] = blockSums[t * 16 + i]; s += loc[i]; }
  tmp[t] = s;
  __syncthreads();
  for (int off = 1; off < 256; off <<= 1) {
    unsigned v = (t >= off) ? tmp[t - off] : 0u;
    __syncthreads();
    tmp[t] += v;
    __syncthreads();
  }
  unsigned excl = tmp[t] - s;
#pragma unroll
  for (int i = 0; i < 16; ++i) { unsigned c = loc[i]; blockSums[t * 16 + i] = excl; excl += c; }
  if (t == 255) *uniqCount = excl;
}

__global__ __launch_bounds__(SCAN_BLOCK)
void gp_scan_pass3(const unsigned* __restrict__ cnt, const unsigned* __restrict__ blockSums,
                   unsigned* __restrict__ rank) {
  __shared__ unsigned tmp[SCAN_BLOCK];
  const int t = threadIdx.x;
  size_t base = (size_t)blockIdx.x * SCAN_TILE + (size_t)t * SCAN_ITEMS;
  unsigned loc[SCAN_ITEMS];
  unsigned s = 0;
#pragma unroll
  for (int i = 0; i < SCAN_ITEMS; ++i) { loc[i] = (cnt[base + i] != 0u) ? 1u : 0u; s += loc[i]; }
  tmp[t] = s;
  __syncthreads();
  for (int off = 1; off < SCAN_BLOCK; off <<= 1) {
    unsigned v = (t >= off) ? tmp[t - off] : 0u;
    __syncthreads();
    tmp[t] += v;
    __syncthreads();
  }
  unsigned excl = tmp[t] - s + blockSums[blockIdx.x];
#pragma unroll
  for (int i = 0; i < SCAN_ITEMS; ++i) { rank[base + i] = excl; excl += loc[i]; }
}

// ------------------------------------------------ decode sorted-unique outputs
__global__ void gp_write_uniq(const unsigned* __restrict__ cnt, const unsigned* __restrict__ rank,
                              int* __restrict__ counts, int* __restrict__ batch_out,
                              int* __restrict__ grid_out) {
  size_t j = (size_t)blockIdx.x * blockDim.x + threadIdx.x;
  if (j >= (size_t)KEYSPACE) return;
  unsigned c = cnt[j];
  if (!c) return;
  unsigned u = rank[j];
  counts[u] = (int)c;
  batch_out[u] = (int)(j >> KEYBITS);
  unsigned rem = (unsigned)j & ((1u << KEYBITS) - 1u);
  grid_out[3 * u]     = (int)(rem >> 14);
  grid_out[3 * u + 1] = (int)((rem >> 7) & 127u);
  grid_out[3 * u + 2] = (int)(rem & 127u);
}

__global__ void gp_write_pad(const unsigned* __restrict__ uniqCount, int* __restrict__ counts,
                             int* __restrict__ batch_out, int* __restrict__ grid_out, int n) {
  int i = blockIdx.x * blockDim.x + threadIdx.x;
  if (i >= n) return;
  if ((unsigned)i >= *uniqCount) {
    counts[i] = 0;
    batch_out[i] = -1;                 // matches -1 // K^3 under floor division
    grid_out[3 * i] = 127; grid_out[3 * i + 1] = 127; grid_out[3 * i + 2] = 127;
  }
}

__global__ void gp_gather_cluster(const int* __restrict__ keys, const unsigned* __restrict__ rank,
                                  int* __restrict__ cluster, int n) {
  int i = blockIdx.x * blockDim.x + threadIdx.x;
  if (i >= n) return;
  cluster[i] = (int)rank[keys[i]];
}

// ----------------------------------------------------------- coord segment sum
__global__ void gp_coord_scatter(const float* __restrict__ coord, const int* __restrict__ cluster,
                                 float* __restrict__ coord_sum, int n) {
  int i = blockIdx.x * blockDim.x + threadIdx.x;
  if (i >= n) return;
  int c = cluster[i];
  atomicAdd(&coord_sum[3 * c],     coord[3 * i]);
  atomicAdd(&coord_sum[3 * c + 1], coord[3 * i + 1]);
  atomicAdd(&coord_sum[3 * c + 2], coord[3 * i + 2]);
}

// ---------------------------------------- WMMA projection + clustered scatter
// Each wave computes a 16x128 tile of feat @ W via 16 chained
// V_WMMA_F32_16X16X4_F32 steps (K=64) x 8 column tiles. A fragments for all 16
// K-steps are hoisted into one unconditional load clause (tail rows clamped to
// n-1; their garbage results are masked at the scatter). Accumulator rows are
// scatter-added into feat_sum[cluster[row]] with fp32 global atomics.
__global__ __launch_bounds__(256)
void gp_proj_wmma(const float* __restrict__ feat, const float* __restrict__ W,
                  const int* __restrict__ cluster, float* __restrict__ feat_sum, int n) {
  // W staged in LDS as interleaved even/odd-k pairs: Wl[kp*COUT+col] = {W[2kp][col], W[2kp+1][col]}
  __shared__ v2f Wl[(CIN / 2) * COUT];              // 32 KB
  const int t = threadIdx.x;
  for (int idx = t; idx < (CIN / 2) * COUT; idx += 256) {
    int kp = idx >> 7, col = idx & (COUT - 1);
    Wl[idx] = v2f{W[(2 * kp) * COUT + col], W[(2 * kp + 1) * COUT + col]};
  }
  __syncthreads();

  const int wave = t >> 5, lane = t & 31;
  const int rowBase = blockIdx.x * 128 + wave * 16;
  const int m = lane & 15;
  const int khalf = lane >> 4;                      // 0: K0/K1 half, 1: K2/K3 half
  const int r = rowBase + m;
  const int rclamp = (r < n) ? r : (n - 1);         // unconditional, in-bounds loads

  // Hoist all A fragments: one clause of 16 global_load_b64.
  const float* arow = feat + (size_t)rclamp * CIN + khalf * 2;
  v2f a[CIN / 4];
#pragma unroll
  for (int kt = 0; kt < CIN / 4; ++kt)
    a[kt] = *(const v2f*)(arow + kt * 4);

  v8f acc[8];
#pragma unroll
  for (int i = 0; i < 8; ++i) acc[i] = v8f{0.f, 0.f, 0.f, 0.f, 0.f, 0.f, 0.f, 0.f};

#pragma unroll
  for (int kt = 0; kt < CIN / 4; ++kt) {
    const int kp = kt * 2 + khalf;
#pragma unroll
    for (int nt = 0; nt < 8; ++nt) {
      v2f b = Wl[kp * COUT + nt * 16 + m];
      acc[nt] = __builtin_amdgcn_wmma_f32_16x16x4_f32(
          false, a[kt], false, b, (short)0, acc[nt], false, false);
    }
  }

  // Per-lane cluster ids for the 8 output rows this half-wave holds.
  int cl[8];
#pragma unroll
  for (int e = 0; e < 8; ++e) {
    int rr = rowBase + e + khalf * 8;
    cl[e] = (rr < n) ? cluster[rr] : -1;
  }
  const int colb = lane & 15;
#pragma unroll
  for (int nt = 0; nt < 8; ++nt) {
#pragma unroll
    for (int e = 0; e < 8; ++e) {
      if (cl[e] >= 0)
        atomicAdd(&feat_sum[(size_t)cl[e] * COUT + nt * 16 + colb], acc[nt][e]);
    }
  }
}

// ------------------------------------------------------------------ finalize
__global__ void gp_final_feat(float* __restrict__ feat_out, const int* __restrict__ counts,
                              const float* __restrict__ bias, int n) {
  size_t i = (size_t)blockIdx.x * blockDim.x + threadIdx.x;
  if (i >= (size_t)n * COUT) return;
  int u = (int)(i >> 7), col = (int)(i & (COUT - 1));
  int c = counts[u];
  float d = (float)((c > 1) ? c : 1);
  float v = feat_out[i] / d;
  if (c > 0) v += bias[col];
  feat_out[i] = v;
}

__global__ void gp_final_coord(float* __restrict__ coord_out, const int* __restrict__ counts, int n) {
  int i = blockIdx.x * blockDim.x + threadIdx.x;
  if (i >= n * 3) return;
  int c = counts[i / 3];
  coord_out[i] /= (float)((c > 1) ? c : 1);
}

// -------------------------------------------------------------------- driver
extern "C" void kernel_launch(void* const* d_in, const int* in_sizes, int n_in,
                              void* d_out, int out_size, void* d_ws, size_t ws_size,
                              hipStream_t stream) {
  const float* feat       = (const float*)d_in[0];
  const float* coord      = (const float*)d_in[1];
  const int*   grid_coord = (const int*)d_in[2];
  const int*   batch      = (const int*)d_in[3];
  const float* W          = (const float*)d_in[4];
  const float* bvec       = (const float*)d_in[5];
  const int n = in_sizes[0] / CIN;

  // Output layout: feat_out[n,128] | coord_out[n,3] | grid_out[n,3] | batch[n] | cluster[n] | counts[n]
  float* feat_out  = (float*)d_out;
  float* coord_out = feat_out + (size_t)n * COUT;
  int*   grid_out  = (int*)(coord_out + (size_t)n * 3);
  int*   batch_out = grid_out + (size_t)n * 3;
  int*   cluster   = batch_out + n;
  int*   counts    = cluster + n;

  // Workspace layout
  char* ws = (char*)d_ws;
  int*      keys      = (int*)ws;       ws += (size_t)n * sizeof(int);
  unsigned* cnt       = (unsigned*)ws;  ws += (size_t)KEYSPACE * sizeof(unsigned);
  unsigned* rank      = (unsigned*)ws;  ws += (size_t)KEYSPACE * sizeof(unsigned);
  unsigned* blockSums = (unsigned*)ws;  ws += (size_t)NUM_SCAN_BLOCKS * sizeof(unsigned);
  unsigned* uniqCount = (unsigned*)ws;

  gp_zero<<<2048, 256, 0, stream>>>(cnt, (size_t)KEYSPACE);
  gp_zero<<<4096, 256, 0, stream>>>((unsigned*)feat_out, (size_t)n * (COUT + 3)); // feat+coord sums

  gp_keys_count<<<(n + 255) / 256, 256, 0, stream>>>(grid_coord, batch, keys, cnt, n);

  gp_scan_pass1<<<NUM_SCAN_BLOCKS, SCAN_BLOCK, 0, stream>>>(cnt, blockSums);
  gp_scan_pass2<<<1, 256, 0, stream>>>(blockSums, uniqCount);
  gp_scan_pass3<<<NUM_SCAN_BLOCKS, SCAN_BLOCK, 0, stream>>>(cnt, blockSums, rank);

  gp_write_uniq<<<(KEYSPACE + 255) / 256, 256, 0, stream>>>(cnt, rank, counts, batch_out, grid_out);
  gp_write_pad<<<(n + 255) / 256, 256, 0, stream>>>(uniqCount, counts, batch_out, grid_out, n);
  gp_gather_cluster<<<(n + 255) / 256, 256, 0, stream>>>(keys, rank, cluster, n);

  gp_coord_scatter<<<(n + 255) / 256, 256, 0, stream>>>(coord, cluster, coord_out, n);
  gp_proj_wmma<<<(n + 127) / 128, 256, 0, stream>>>(feat, W, cluster, feat_out, n);

  gp_final_feat<<<(int)(((size_t)n * COUT + 255) / 256), 256, 0, stream>>>(feat_out, counts, bvec, n);
  gp_final_coord<<<(n * 3 + 255) / 256, 256, 0, stream>>>(coord_out, counts, n);
}